// GCNEncoder_24137716204247
// MI455X (gfx1250) — compile-verified
//
#include <hip/hip_runtime.h>

typedef float v2f __attribute__((ext_vector_type(2)));
typedef float v8f __attribute__((ext_vector_type(8)));

#define IN_C      256
#define HID_C     128
#define OUT_C     64

// ---------------------------------------------------------------------------
// helpers
// ---------------------------------------------------------------------------
__device__ __forceinline__ void atomic_add_f32(float* p, float v) {
    unsafeAtomicAdd(p, v);   // hardware global_atomic_add_f32
}

__global__ __launch_bounds__(256) void k_zero4(float4* __restrict__ p, int n4) {
    int i = blockIdx.x * 256 + threadIdx.x;
    if (i < n4) p[i] = make_float4(0.f, 0.f, 0.f, 0.f);
}

// Pack W[K,N] into pair-major Wp[K/2][N] of float2 {W[k,n], W[k+1,n]} so the
// WMMA B-fragment load is a single coalesced global_load_b64.
__global__ __launch_bounds__(256) void k_packw(const float* __restrict__ W,
                                               float2* __restrict__ Wp,
                                               int K, int N) {
    int i = blockIdx.x * 256 + threadIdx.x;     // over (K/2)*N
    if (i >= (K / 2) * N) return;
    int kp = i / N, n = i % N;
    Wp[i] = make_float2(W[(2 * kp) * N + n], W[(2 * kp + 1) * N + n]);
}

// ---------------------------------------------------------------------------
// degree + normalization
// ---------------------------------------------------------------------------
__global__ __launch_bounds__(256) void k_degree(const int* __restrict__ dst,
                                                float* __restrict__ deg, int E) {
    int e = blockIdx.x * 256 + threadIdx.x;
    if (e < E) atomic_add_f32(&deg[dst[e]], 1.0f);
}

__global__ __launch_bounds__(256) void k_dinv(float* __restrict__ deg_dinv, int n) {
    int i = blockIdx.x * 256 + threadIdx.x;
    if (i < n) deg_dinv[i] = rsqrtf(deg_dinv[i] + 1.0f);  // deg includes self-loop
}

// ---------------------------------------------------------------------------
// GEMM via V_WMMA_F32_16X16X4_F32 (fragment layouts per ISA 7.12.2).
// Block owns 16 output rows (N_NODES=50000 is a multiple of 16 -> no guards ->
// EXEC all ones as WMMA requires); each wave owns a 16-col N tile.
// ---------------------------------------------------------------------------
template <int K, int N>
__device__ __forceinline__ void gemm_tile(const float* __restrict__ X,
                                          const float2* __restrict__ Wp,
                                          float* __restrict__ H) {
    const int lane = threadIdx.x & 31;
    const int wave = threadIdx.x >> 5;
    const int m0   = blockIdx.x * 16;
    const int n0   = wave * 16;
    const int half = lane >> 4;              // 0: K{0,1}, 1: K{2,3}
    const int l15  = lane & 15;
    const int koff = half * 2;

    const float*  __restrict__ xrow = X + (size_t)(m0 + l15) * K + koff;
    const float2* __restrict__ wrow = Wp + (size_t)half * N + n0 + l15;

    v8f c = {};
    #pragma unroll 4
    for (int k = 0; k < K; k += 4) {
        float2 av = *(const float2*)(xrow + k);          // global_load_b64
        float2 bv = wrow[(size_t)(k >> 1) * N];          // global_load_b64
        v2f a = {av.x, av.y};
        v2f b = {bv.x, bv.y};
        c = __builtin_amdgcn_wmma_f32_16x16x4_f32(false, a, false, b,
                                                  (short)0, c, false, false);
    }
    #pragma unroll
    for (int v = 0; v < 8; ++v)
        H[(size_t)(m0 + v + 8 * half) * N + n0 + l15] = c[v];
}

__global__ __launch_bounds__(256) void k_gemm1(const float* __restrict__ X,
                                               const float2* __restrict__ Wp,
                                               float* __restrict__ H) {
    gemm_tile<IN_C, HID_C>(X, Wp, H);        // 8 waves cover 128 cols
}

__global__ __launch_bounds__(128) void k_gemm2(const float* __restrict__ X,
                                               const float2* __restrict__ Wp,
                                               float* __restrict__ H) {
    gemm_tile<HID_C, OUT_C>(X, Wp, H);       // 4 waves cover 64 cols
}

// ---------------------------------------------------------------------------
// Edge scatter: one wave32 per edge; lane owns 4 (or 2) consecutive features:
// one global_load_b128 gather + coalesced f32 atomics. 256 thr = 8 edges/blk.
// ---------------------------------------------------------------------------
__global__ __launch_bounds__(256) void k_scatter128(const float* __restrict__ H,
                                                    const int* __restrict__ src,
                                                    const int* __restrict__ dst,
                                                    const float* __restrict__ dinv,
                                                    float* __restrict__ agg, int E) {
    int e = blockIdx.x * 8 + (threadIdx.x >> 5);
    if (e >= E) return;
    const int lane = threadIdx.x & 31;
    const int s = src[e], d = dst[e];
    const float norm = dinv[s] * dinv[d];
    const float4 hv = *(const float4*)(H + (size_t)s * HID_C + lane * 4);
    float* __restrict__ ad = agg + (size_t)d * HID_C + lane * 4;
    atomic_add_f32(ad + 0, hv.x * norm);
    atomic_add_f32(ad + 1, hv.y * norm);
    atomic_add_f32(ad + 2, hv.z * norm);
    atomic_add_f32(ad + 3, hv.w * norm);
}

__global__ __launch_bounds__(256) void k_scatter64(const float* __restrict__ H,
                                                   const int* __restrict__ src,
                                                   const int* __restrict__ dst,
                                                   const float* __restrict__ dinv,
                                                   float* __restrict__ agg, int E) {
    int e = blockIdx.x * 8 + (threadIdx.x >> 5);
    if (e >= E) return;
    const int lane = threadIdx.x & 31;
    const int s = src[e], d = dst[e];
    const float norm = dinv[s] * dinv[d];
    const float2 hv = *(const float2*)(H + (size_t)s * OUT_C + lane * 2);
    float* __restrict__ ad = agg + (size_t)d * OUT_C + lane * 2;
    atomic_add_f32(ad + 0, hv.x * norm);
    atomic_add_f32(ad + 1, hv.y * norm);
}

// ---------------------------------------------------------------------------
// self-loop + bias (+ ReLU for layer 1), float4-vectorized, in place on agg
// ---------------------------------------------------------------------------
__global__ __launch_bounds__(256) void k_finish1(float4* __restrict__ agg,
                                                 const float4* __restrict__ H,
                                                 const float* __restrict__ dinv,
                                                 const float4* __restrict__ bias,
                                                 int total4) {
    int idx = blockIdx.x * 256 + threadIdx.x;
    if (idx >= total4) return;
    int i = idx >> 5;                 // 4*idx / 128
    int f = idx & 31;                 // float4 column
    float di = dinv[i]; di *= di;
    float4 a = agg[idx], h = H[idx], b = bias[f];
    a.x = fmaxf(a.x + h.x * di + b.x, 0.f);
    a.y = fmaxf(a.y + h.y * di + b.y, 0.f);
    a.z = fmaxf(a.z + h.z * di + b.z, 0.f);
    a.w = fmaxf(a.w + h.w * di + b.w, 0.f);
    agg[idx] = a;
}

__global__ __launch_bounds__(256) void k_finish2(float4* __restrict__ out,
                                                 const float4* __restrict__ H,
                                                 const float* __restrict__ dinv,
                                                 const float4* __restrict__ bias,
                                                 int total4) {
    int idx = blockIdx.x * 256 + threadIdx.x;
    if (idx >= total4) return;
    int i = idx >> 4;                 // 4*idx / 64
    int f = idx & 15;
    float di = dinv[i]; di *= di;
    float4 a = out[idx], h = H[idx], b = bias[f];
    a.x += h.x * di + b.x;
    a.y += h.y * di + b.y;
    a.z += h.z * di + b.z;
    a.w += h.w * di + b.w;
    out[idx] = a;
}

// ---------------------------------------------------------------------------
// launch
// ---------------------------------------------------------------------------
extern "C" void kernel_launch(void* const* d_in, const int* in_sizes, int n_in,
                              void* d_out, int out_size, void* d_ws, size_t ws_size,
                              hipStream_t stream) {
    const float* x   = (const float*)d_in[0];
    const float* W1  = (const float*)d_in[1];
    const float* b1  = (const float*)d_in[2];
    const float* W2  = (const float*)d_in[3];
    const float* b2  = (const float*)d_in[4];
    const int*   ei  = (const int*)d_in[5];

    const int E = in_sizes[5] / 2;
    const int* srcp = ei;
    const int* dstp = ei + E;

    const int n = in_sizes[0] / IN_C;          // 50000 (multiple of 16)

    // workspace layout (bytes)
    char* ws = (char*)d_ws;
    float*  dinv = (float*)(ws);                               // n floats
    float2* wp1  = (float2*)(ws + 0x40000);                    // 128*128 float2 = 128KB
    float2* wp2  = (float2*)(ws + 0x60000);                    // 64*64  float2 =  32KB
    float*  h1   = (float*)(ws + 0x80000);                     // n*128 f32
    float*  agg1 = (float*)(ws + 0x80000 + (size_t)n * HID_C * 4);
    float*  h2   = (float*)(ws + 0x80000 + 2 * (size_t)n * HID_C * 4);
    float*  out  = (float*)d_out;                              // n*64, scatter target

    const int t1 = n * HID_C;   // 6,400,000
    const int t2 = n * OUT_C;   // 3,200,000

    // zero accumulation targets (harness does not re-zero between replays)
    k_zero4<<<(n / 4 + 255) / 256, 256, 0, stream>>>((float4*)dinv, n / 4);
    k_zero4<<<(t1 / 4 + 255) / 256, 256, 0, stream>>>((float4*)agg1, t1 / 4);
    k_zero4<<<(t2 / 4 + 255) / 256, 256, 0, stream>>>((float4*)out, t2 / 4);

    // pack weights into WMMA-B pair-major order
    k_packw<<<((IN_C / 2) * HID_C + 255) / 256, 256, 0, stream>>>(W1, wp1, IN_C, HID_C);
    k_packw<<<((HID_C / 2) * OUT_C + 255) / 256, 256, 0, stream>>>(W2, wp2, HID_C, OUT_C);

    // degree + normalization
    k_degree<<<(E + 255) / 256, 256, 0, stream>>>(dstp, dinv, E);
    k_dinv<<<(n + 255) / 256, 256, 0, stream>>>(dinv, n);

    // layer 1
    k_gemm1<<<n / 16, 256, 0, stream>>>(x, wp1, h1);
    k_scatter128<<<(E + 7) / 8, 256, 0, stream>>>(h1, srcp, dstp, dinv, agg1, E);
    k_finish1<<<(t1 / 4 + 255) / 256, 256, 0, stream>>>((float4*)agg1, (const float4*)h1,
                                                        dinv, (const float4*)b1, t1 / 4);

    // layer 2
    k_gemm2<<<n / 16, 128, 0, stream>>>(agg1, wp2, h2);
    k_scatter64<<<(E + 7) / 8, 256, 0, stream>>>(h2, srcp, dstp, dinv, out, E);
    k_finish2<<<(t2 / 4 + 255) / 256, 256, 0, stream>>>((float4*)out, (const float4*)h2,
                                                        dinv, (const float4*)b2, t2 / 4);
}